// model_29549374997210
// MI455X (gfx1250) — compile-verified
//
#include <hip/hip_runtime.h>
#include <hip/hip_bf16.h>

typedef __attribute__((ext_vector_type(2))) float v2f;
typedef __attribute__((ext_vector_type(8))) float v8f;

static __device__ inline v8f wmma4(v2f a, v2f b, v8f c) {
  // D = A(16x4 f32) * B(4x16 f32) + C(16x16 f32)
  return __builtin_amdgcn_wmma_f32_16x16x4_f32(false, a, false, b, (short)0, c,
                                               false, false);
}

// ---------------------------------------------------------------------------
// Generic batched complex GEMM via V_WMMA_F32_16X16X4_F32, 2x2 register tiled.
// C[i,j] = beta*C[i,j] + alpha * sum_t A[i*lda+t] * (conjB? conj(Bel) : Bel)
// where Bel = B[j*bsj + (t/TD)*bsq + (t%TD)*bsr].
// One wave per 32x32 complex tile (4 sub-tiles of 16x16): per k-step we load
// 2 A fragments + 2 B fragments (real+imag) and issue 16 WMMAs -> 2x the
// arithmetic intensity of a 16x16 tile and 16 independent WMMA chains of ILP.
// M,N multiples of 32; K multiple of 4.
// ---------------------------------------------------------------------------
__global__ __launch_bounds__(32)
void cgemm32(const float* __restrict__ Ar, const float* __restrict__ Ai,
             long lda, long strideA,
             const float* __restrict__ Br, const float* __restrict__ Bi,
             long bsj, long bsq, long bsr, int TD, long strideB,
             float* __restrict__ Cr, float* __restrict__ Ci,
             long ldc, long strideC,
             int K, int conjB, float alpha, float beta)
{
  const int lane = threadIdx.x;
  const int tm = blockIdx.x, tn = blockIdx.y, b = blockIdx.z;
  const float* ar = Ar + (long)b * strideA;
  const float* ai = Ai + (long)b * strideA;
  const float* br = Br + (long)b * strideB;
  const float* bi = Bi + (long)b * strideB;

  const int arow  = tm * 32 + (lane & 15);
  const int bcol  = tn * 32 + (lane & 15);
  const int khalf = (lane >> 4) << 1;   // 0 for lanes 0-15, 2 for lanes 16-31

  long aoff = (long)arow * lda + khalf;
  long boff = (long)bcol * bsj + (long)khalf * bsr;
  const long arow_step = 16 * lda;      // A fragment row-block stride
  const long bcol_step = 16 * bsj;      // B fragment col-block stride
  int  r    = khalf;

  v8f acc_rr[2][2], acc_ii[2][2], acc_ri[2][2], acc_ir[2][2];
#pragma unroll
  for (int mi = 0; mi < 2; ++mi)
#pragma unroll
    for (int ni = 0; ni < 2; ++ni) {
      v8f z = {};
      acc_rr[mi][ni] = z; acc_ii[mi][ni] = z;
      acc_ri[mi][ni] = z; acc_ir[mi][ni] = z;
    }

  for (int k = 0; k < K; k += 4) {
    v2f a_r[2], a_i[2], b_r[2], b_i[2];
#pragma unroll
    for (int mi = 0; mi < 2; ++mi) {
      long o = aoff + mi * arow_step;
      a_r[mi].x = ar[o]; a_r[mi].y = ar[o + 1];
      a_i[mi].x = ai[o]; a_i[mi].y = ai[o + 1];
    }
#pragma unroll
    for (int ni = 0; ni < 2; ++ni) {
      long o = boff + ni * bcol_step;
      b_r[ni].x = br[o]; b_r[ni].y = br[o + bsr];
      b_i[ni].x = bi[o]; b_i[ni].y = bi[o + bsr];
    }
#pragma unroll
    for (int mi = 0; mi < 2; ++mi)
#pragma unroll
      for (int ni = 0; ni < 2; ++ni) {
        acc_rr[mi][ni] = wmma4(a_r[mi], b_r[ni], acc_rr[mi][ni]);
        acc_ii[mi][ni] = wmma4(a_i[mi], b_i[ni], acc_ii[mi][ni]);
        acc_ri[mi][ni] = wmma4(a_r[mi], b_i[ni], acc_ri[mi][ni]);
        acc_ir[mi][ni] = wmma4(a_i[mi], b_r[ni], acc_ir[mi][ni]);
      }
    aoff += 4;
    boff += 4 * bsr;
    r += 4;
    if (r >= TD) { r -= TD; boff += bsq - (long)TD * bsr; }  // uniform per wave
  }

  float* cr = Cr + (long)b * strideC;
  float* ci = Ci + (long)b * strideC;
  const int crow_base = tm * 32 + ((lane >> 4) << 3);
  const int ccol_base = tn * 32 + (lane & 15);
#pragma unroll
  for (int mi = 0; mi < 2; ++mi)
#pragma unroll
    for (int ni = 0; ni < 2; ++ni) {
      v8f vr, vi;
      if (conjB) {
        vr = acc_rr[mi][ni] + acc_ii[mi][ni];
        vi = acc_ir[mi][ni] - acc_ri[mi][ni];
      } else {
        vr = acc_rr[mi][ni] - acc_ii[mi][ni];
        vi = acc_ri[mi][ni] + acc_ir[mi][ni];
      }
      const int crow0 = crow_base + mi * 16;
      const int ccol  = ccol_base + ni * 16;
#pragma unroll
      for (int v = 0; v < 8; ++v) {
        long off = (long)(crow0 + v) * ldc + ccol;
        float pr = 0.f, pi = 0.f;
        if (beta != 0.f) { pr = cr[off]; pi = ci[off]; }
        cr[off] = pr + alpha * vr[v];
        ci[off] = pi + alpha * vi[v];
      }
    }
}

// ---------------------------------------------------------------------------
// T1[m,u,k] = sum_n conj(VU[n,m]) * UU[n,u,k]; stored [m*1024 + u*128 + k]
// ---------------------------------------------------------------------------
__global__ void k_t1(const float* __restrict__ VU, const float* __restrict__ UU,
                     float* __restrict__ T1) {
  const int k = blockIdx.x;
  const float* VUr = VU;  const float* VUi = VU + 12288;
  const float* UUr = UU;  const float* UUi = UU + 131072;
  float* T1r = T1;        float* T1i = T1 + 98304;
  for (int e = threadIdx.x; e < 768; e += blockDim.x) {
    int m = e >> 3, u = e & 7;
    float sr = 0.f, si = 0.f;
    for (int n = 0; n < 128; ++n) {
      float a_r = VUr[n * 96 + m], a_i = VUi[n * 96 + m];
      float b_r = UUr[(n * 8 + u) * 128 + k], b_i = UUi[(n * 8 + u) * 128 + k];
      sr += a_r * b_r + a_i * b_i;   // conj(a)*b
      si += a_r * b_i - a_i * b_r;
    }
    T1r[m * 1024 + u * 128 + k] = sr;
    T1i[m * 1024 + u * 128 + k] = si;
  }
}

// T[m,d,l] = sum_n conj(G[n,m,l]) * UD[n,d,l]   (G: 16x96x128, UD: 16x8x128)
__global__ void k_tdl(const float* __restrict__ G, const float* __restrict__ UD,
                      float* __restrict__ T) {
  const int l = blockIdx.x;
  const float* Gr = G;   const float* Gi = G + 196608;
  const float* Ur = UD;  const float* Ui = UD + 16384;
  float* Tr = T;         float* Ti = T + 98304;
  for (int e = threadIdx.x; e < 768; e += blockDim.x) {
    int m = e >> 3, d = e & 7;
    float sr = 0.f, si = 0.f;
    for (int n = 0; n < 16; ++n) {
      float a_r = Gr[(n * 96 + m) * 128 + l], a_i = Gi[(n * 96 + m) * 128 + l];
      float b_r = Ur[(n * 8 + d) * 128 + l],  b_i = Ui[(n * 8 + d) * 128 + l];
      sr += a_r * b_r + a_i * b_i;
      si += a_r * b_i - a_i * b_r;
    }
    Tr[m * 1024 + d * 128 + l] = sr;
    Ti[m * 1024 + d * 128 + l] = si;
  }
}

// PPH_k (LDS) then Q[a,f,k] = sum_b PPH[a,b,k] conj(GU[f,b,k]);
// stored Qmat[f*2048 + a*128 + k]
__global__ void k_pphq(const float* __restrict__ P, const float* __restrict__ GU,
                       float* __restrict__ Q) {
  __shared__ float pr[256], pi[256];
  const int k = blockIdx.x;
  const float* Pr = P;   const float* Pi = P + 16384;
  const float* Gr = GU;  const float* Gi = GU + 196608;
  float* Qr = Q;         float* Qi = Q + 196608;
  for (int e = threadIdx.x; e < 256; e += blockDim.x) {
    int a = e >> 4, b2 = e & 15;
    float sr = 0.f, si = 0.f;
    for (int u = 0; u < 8; ++u) {
      float a_r = Pr[(a * 8 + u) * 128 + k],  a_i = Pi[(a * 8 + u) * 128 + k];
      float b_r = Pr[(b2 * 8 + u) * 128 + k], b_i = Pi[(b2 * 8 + u) * 128 + k];
      sr += a_r * b_r + a_i * b_i;   // a * conj(b)
      si += a_i * b_r - a_r * b_i;
    }
    pr[e] = sr; pi[e] = si;
  }
  __syncthreads();
  for (int e = threadIdx.x; e < 1536; e += blockDim.x) {
    int a = e / 96, f = e - a * 96;
    float sr = 0.f, si = 0.f;
    for (int b2 = 0; b2 < 16; ++b2) {
      float x_r = pr[a * 16 + b2], x_i = pi[a * 16 + b2];
      float g_r = Gr[f * 2048 + b2 * 128 + k], g_i = Gi[f * 2048 + b2 * 128 + k];
      sr += x_r * g_r + x_i * g_i;   // x * conj(g)
      si += x_i * g_r - x_r * g_i;
    }
    Qr[f * 2048 + a * 128 + k] = sr;
    Qi[f * 2048 + a * 128 + k] = si;
  }
}

// Per-k chain: M1 = T1_k WU_k (global); X1 = T1_k WU_k^H (LDS);
// Y[m,a,k] = sum_v X1[m,v] conj(P[a,v,k])  stored [m*2048 + a*128 + k]
__global__ void k_chainU(const float* __restrict__ T1, const float* __restrict__ WU,
                         const float* __restrict__ P,
                         float* __restrict__ M1, float* __restrict__ Y) {
  __shared__ float xr[768], xi[768];
  const int k = blockIdx.x;
  const float* Tr = T1;  const float* Ti = T1 + 98304;
  const float* Wr = WU;  const float* Wi = WU + 8192;
  const float* Pr = P;   const float* Pi = P + 16384;
  float* M1r = M1;       float* M1i = M1 + 98304;
  float* Yr  = Y;        float* Yi  = Y + 196608;
  for (int e = threadIdx.x; e < 768; e += blockDim.x) {
    int m = e >> 3, c = e & 7;
    float mr = 0.f, mi = 0.f, sxr = 0.f, sxi = 0.f;
    for (int v = 0; v < 8; ++v) {
      float t_r = Tr[m * 1024 + v * 128 + k], t_i = Ti[m * 1024 + v * 128 + k];
      float w_r = Wr[(v * 8 + c) * 128 + k], w_i = Wi[(v * 8 + c) * 128 + k];
      mr += t_r * w_r - t_i * w_i;             // T1[m,v] * WU[v,c]
      mi += t_r * w_i + t_i * w_r;
      float u_r = Wr[(c * 8 + v) * 128 + k], u_i = Wi[(c * 8 + v) * 128 + k];
      sxr += t_r * u_r + t_i * u_i;            // T1[m,v] * conj(WU[c,v])
      sxi += t_i * u_r - t_r * u_i;
    }
    M1r[m * 1024 + c * 128 + k] = mr;
    M1i[m * 1024 + c * 128 + k] = mi;
    xr[e] = sxr; xi[e] = sxi;
  }
  __syncthreads();
  for (int e = threadIdx.x; e < 1536; e += blockDim.x) {
    int m = e >> 4, a = e & 15;
    float sr = 0.f, si = 0.f;
    for (int v = 0; v < 8; ++v) {
      float x_r = xr[m * 8 + v], x_i = xi[m * 8 + v];
      float p_r = Pr[(a * 8 + v) * 128 + k], p_i = Pi[(a * 8 + v) * 128 + k];
      sr += x_r * p_r + x_i * p_i;   // X1 * conj(P)
      si += x_i * p_r - x_r * p_i;
    }
    Yr[m * 2048 + a * 128 + k] = sr;
    Yi[m * 2048 + a * 128 + k] = si;
  }
}

// Per-l chain: M = T_l WD_l (global+LDS); optional X = T_l WD_l^H (global);
// V[m,n,l] = sum_e M[m,e] conj(UD[n,e,l])  stored [m*2048 + n*128 + l]
__global__ void k_chainD(const float* __restrict__ T, const float* __restrict__ WD,
                         const float* __restrict__ UD,
                         float* __restrict__ M, float* __restrict__ X,
                         float* __restrict__ V, int doX) {
  __shared__ float mr_s[768], mi_s[768];
  const int l = blockIdx.x;
  const float* Tr = T;   const float* Ti = T + 98304;
  const float* Wr = WD;  const float* Wi = WD + 8192;
  const float* Ur = UD;  const float* Ui = UD + 16384;
  float* Mr = M;         float* Mi = M + 98304;
  float* Vr = V;         float* Vi = V + 196608;
  for (int e = threadIdx.x; e < 768; e += blockDim.x) {
    int m = e >> 3, c = e & 7;
    float mr = 0.f, mi = 0.f, sxr = 0.f, sxi = 0.f;
    for (int d = 0; d < 8; ++d) {
      float t_r = Tr[m * 1024 + d * 128 + l], t_i = Ti[m * 1024 + d * 128 + l];
      float w_r = Wr[(d * 8 + c) * 128 + l], w_i = Wi[(d * 8 + c) * 128 + l];
      mr += t_r * w_r - t_i * w_i;             // T[m,d] * WD[d,c]
      mi += t_r * w_i + t_i * w_r;
      if (doX) {
        float u_r = Wr[(c * 8 + d) * 128 + l], u_i = Wi[(c * 8 + d) * 128 + l];
        sxr += t_r * u_r + t_i * u_i;          // T[m,d] * conj(WD[c,d])
        sxi += t_i * u_r - t_r * u_i;
      }
    }
    Mr[m * 1024 + c * 128 + l] = mr;
    Mi[m * 1024 + c * 128 + l] = mi;
    mr_s[e] = mr; mi_s[e] = mi;
    if (doX) {
      X[m * 1024 + c * 128 + l]         = sxr;
      X[98304 + m * 1024 + c * 128 + l] = sxi;
    }
  }
  __syncthreads();
  for (int e = threadIdx.x; e < 1536; e += blockDim.x) {
    int m = e >> 4, n = e & 15;
    float sr = 0.f, si = 0.f;
    for (int d = 0; d < 8; ++d) {
      float a_r = mr_s[m * 8 + d], a_i = mi_s[m * 8 + d];
      float u_r = Ur[(n * 8 + d) * 128 + l], u_i = Ui[(n * 8 + d) * 128 + l];
      sr += a_r * u_r + a_i * u_i;   // M * conj(UD)
      si += a_i * u_r - a_r * u_i;
    }
    Vr[m * 2048 + n * 128 + l] = sr;
    Vi[m * 2048 + n * 128 + l] = si;
  }
}

__global__ void k_copy(const float* __restrict__ src, float* __restrict__ dst, int n) {
  int i = blockIdx.x * blockDim.x + threadIdx.x;
  if (i < n) dst[i] = src[i];
}

// stride-2 2x2 VALID conv + bias + relu; NCHW, OIHW
__global__ void k_conv(const float* __restrict__ in, const float* __restrict__ w,
                       const float* __restrict__ bias, float* __restrict__ out,
                       int IC, int OC, int IS, int OS) {
  int idx = blockIdx.x * blockDim.x + threadIdx.x;
  int total = OC * OS * OS;
  if (idx >= total) return;
  int o = idx / (OS * OS);
  int rem = idx - o * OS * OS;
  int y = rem / OS;
  int x = rem - y * OS;
  float s = bias[o];
  for (int i = 0; i < IC; ++i)
    for (int ky = 0; ky < 2; ++ky)
      for (int kx = 0; kx < 2; ++kx)
        s += w[((o * IC + i) * 2 + ky) * 2 + kx] *
             in[(i * IS + (2 * y + ky)) * IS + (2 * x + kx)];
  out[idx] = fmaxf(s, 0.f);
}

__global__ void k_fc(const float* __restrict__ in, const float* __restrict__ w,
                     const float* __restrict__ bias, float* __restrict__ out,
                     int IN, int ON) {
  int o = blockIdx.x * blockDim.x + threadIdx.x;
  if (o >= ON) return;
  float s = bias[o];
  for (int i = 0; i < IN; ++i) s += w[(long)o * IN + i] * in[i];
  out[o] = fmaxf(s, 0.f);
}

// ---------------------------------------------------------------------------
// Workspace layout (float offsets).  Total ~3.70M floats (~14.8 MB).
// ---------------------------------------------------------------------------
static const size_t OFF_T1  = 0;         // 196608  [m*1024+u*128+k], r then i
static const size_t OFF_T2  = 196608;    // 196608
static const size_t OFF_T3  = 393216;    // 196608
static const size_t OFF_Q   = 589824;    // 393216  Qmat[f*2048 + a*128 + k]
static const size_t OFF_M1  = 983040;    // 196608
static const size_t OFF_Y   = 1179648;   // 393216  [m*2048 + a*128 + k]
static const size_t OFF_M2  = 1572864;   // 196608
static const size_t OFF_X2  = 1769472;   // 196608
static const size_t OFF_V2  = 1966080;   // 393216  [m*2048 + n*128 + l]
static const size_t OFF_M3  = 2359296;   // 196608
static const size_t OFF_TL  = 2555904;   // 393216  [m*2048 + n*128 + l]
static const size_t OFF_R   = 2949120;   // 393216  [n*12288 + f*128 + l]
static const size_t OFF_B11 = 3342336;   // 24576   (96x128)
static const size_t OFF_B12 = 3366912;   // 24576   (128x96)
static const size_t OFF_B21 = 3391488;   // 24576   (96x128)
static const size_t OFF_B22 = 3416064;   // 24576   (128x96)
static const size_t OFF_G   = 3440640;   // 24576   (96x128)
static const size_t OFF_FFH = 3465216;   // 32768   (128x128)
static const size_t OFF_X   = 3497984;   // 165888  9 matrices x [re|im] x 9216
static const size_t OFF_C1O = 3663872;   // 27648
static const size_t OFF_C2O = 3691520;   // 4608
static const size_t OFF_C3O = 3696128;   // 864
static const size_t OFF_F1  = 3696992;   // 1000
static const size_t OFF_F2  = 3697992;   // 700

extern "C" void kernel_launch(void* const* d_in, const int* in_sizes, int n_in,
                              void* d_out, int out_size, void* d_ws, size_t ws_size,
                              hipStream_t stream) {
  (void)in_sizes; (void)n_in; (void)out_size; (void)ws_size;
  const float* VU = (const float*)d_in[0];
  const float* UU = (const float*)d_in[1];
  const float* WU = (const float*)d_in[2];
  const float* GD = (const float*)d_in[3];
  const float* UD = (const float*)d_in[4];
  const float* WD = (const float*)d_in[5];
  const float* Z  = (const float*)d_in[6];
  const float* HU = (const float*)d_in[7];
  const float* HD = (const float*)d_in[8];
  const float* P  = (const float*)d_in[9];
  const float* GU = (const float*)d_in[10];
  const float* F  = (const float*)d_in[11];
  const float* VD = (const float*)d_in[12];
  const float* J  = (const float*)d_in[13];
  const float* c1w = (const float*)d_in[14]; const float* c1b = (const float*)d_in[15];
  const float* c2w = (const float*)d_in[16]; const float* c2b = (const float*)d_in[17];
  const float* c3w = (const float*)d_in[18]; const float* c3b = (const float*)d_in[19];
  const float* f1w = (const float*)d_in[20]; const float* f1b = (const float*)d_in[21];
  const float* f2w = (const float*)d_in[22]; const float* f2b = (const float*)d_in[23];
  const float* f3w = (const float*)d_in[24]; const float* f3b = (const float*)d_in[25];
  float* ws = (float*)d_ws;
  float* X  = ws + OFF_X;

  // ---- stage 1: per-slice precomputations --------------------------------
  k_t1  <<<128, 256, 0, stream>>>(VU, UU, ws + OFF_T1);
  k_tdl <<<128, 256, 0, stream>>>(GD, UD, ws + OFF_T2);
  k_tdl <<<128, 256, 0, stream>>>(Z,  UD, ws + OFF_T3);
  k_pphq<<<128, 256, 0, stream>>>(P, GU, ws + OFF_Q);
  k_chainU<<<128, 256, 0, stream>>>(ws + OFF_T1, WU, P, ws + OFF_M1, ws + OFF_Y);
  k_chainD<<<128, 256, 0, stream>>>(ws + OFF_T2, WD, UD,
                                    ws + OFF_M2, ws + OFF_X2, ws + OFF_V2, 1);
  k_chainD<<<128, 256, 0, stream>>>(ws + OFF_T3, WD, UD,
                                    ws + OFF_M3, ws + OFF_X2, ws + OFF_TL, 0);

  // ---- stage 2: everything GEMM-shaped goes through the WMMA kernel ------
  auto cg = [&](const float* A, long aI, long lda, long sA,
                const float* B, long bI, long bsj, long bsq, long bsr, int TD, long sB,
                float* C, long cI, long ldc, long sC,
                int M, int N, int K, int conjB, float alpha, float beta, int batch) {
    dim3 g(M / 32, N / 32, batch);
    cgemm32<<<g, 32, 0, stream>>>(A, A + aI, lda, sA, B, B + bI, bsj, bsq, bsr, TD, sB,
                                  C, C + cI, ldc, sC, K, conjB, alpha, beta);
  };

  // A1 = M1 @ T1^H                                    -> slot 0
  cg(ws + OFF_M1, 98304, 1024, 0, ws + OFF_T1, 98304, 1024, 0, 1, 1024, 0,
     X + 0 * 18432, 9216, 96, 0, 96, 96, 1024, 1, 1.f, 0.f, 1);
  // B1_1 = M1 @ UU^H   (UU row n contiguous over (u,k))
  cg(ws + OFF_M1, 98304, 1024, 0, UU, 131072, 1024, 0, 1, 1024, 0,
     ws + OFF_B11, 12288, 128, 0, 96, 128, 1024, 1, 1.f, 0.f, 1);
  // B1_3 = Y @ GU^H                                   -> slot 1 (B1)
  cg(ws + OFF_Y, 196608, 2048, 0, GU, 196608, 2048, 0, 1, 2048, 0,
     X + 1 * 18432, 9216, 96, 0, 96, 96, 2048, 1, 1.f, 0.f, 1);
  // B1_2 = HU @ Q^T (no conj)
  cg(HU, 262144, 2048, 0, ws + OFF_Q, 196608, 2048, 0, 1, 2048, 0,
     ws + OFF_B12, 12288, 96, 0, 128, 96, 2048, 0, 1.f, 0.f, 1);
  // B1 -= B1_1 @ B1_2   (B1_2 is [n][f]: bsj=1, bsr=96)
  cg(ws + OFF_B11, 12288, 128, 0, ws + OFF_B12, 12288, 1, 0, 96, 128, 0,
     X + 1 * 18432, 9216, 96, 0, 96, 96, 128, 0, -1.f, 1.f, 1);
  // C1 = GU @ Q^T                                     -> slot 2
  cg(GU, 196608, 2048, 0, ws + OFF_Q, 196608, 2048, 0, 1, 2048, 0,
     X + 2 * 18432, 9216, 96, 0, 96, 96, 2048, 0, 1.f, 0.f, 1);
  // A2 = M2 @ T2^H                                    -> slot 3
  cg(ws + OFF_M2, 98304, 1024, 0, ws + OFF_T2, 98304, 1024, 0, 1, 1024, 0,
     X + 3 * 18432, 9216, 96, 0, 96, 96, 1024, 1, 1.f, 0.f, 1);
  // FFH = F @ F^H
  cg(F, 131072, 1024, 0, F, 131072, 1024, 0, 1, 1024, 0,
     ws + OFF_FFH, 16384, 128, 0, 128, 128, 1024, 1, 1.f, 0.f, 1);
  // B2_2 = FFH @ VD^H
  cg(ws + OFF_FFH, 16384, 128, 0, VD, 12288, 128, 0, 1, 128, 0,
     ws + OFF_B22, 12288, 96, 0, 128, 96, 128, 1, 1.f, 0.f, 1);
  // G = X2 @ F^H
  cg(ws + OFF_X2, 98304, 1024, 0, F, 131072, 1024, 0, 1, 1024, 0,
     ws + OFF_G, 12288, 128, 0, 96, 128, 1024, 1, 1.f, 0.f, 1);
  // B2_3 = G @ VD^H                                   -> slot 4 (B2)
  cg(ws + OFF_G, 12288, 128, 0, VD, 12288, 128, 0, 1, 128, 0,
     X + 4 * 18432, 9216, 96, 0, 96, 96, 128, 1, 1.f, 0.f, 1);
  // B2_1 = V2 @ HD   (HD[n,t,l]: bsj=128, bsq=16384, bsr=1, TD=128)
  cg(ws + OFF_V2, 196608, 2048, 0, HD, 262144, 128, 16384, 1, 128, 0,
     ws + OFF_B21, 12288, 128, 0, 96, 128, 2048, 0, 1.f, 0.f, 1);
  // B2 -= B2_1 @ B2_2
  cg(ws + OFF_B21, 12288, 128, 0, ws + OFF_B22, 12288, 1, 0, 96, 128, 0,
     X + 4 * 18432, 9216, 96, 0, 96, 96, 128, 0, -1.f, 1.f, 1);
  // C2 = VD @ B2_2                                    -> slot 5
  cg(VD, 12288, 128, 0, ws + OFF_B22, 12288, 1, 0, 96, 128, 0,
     X + 5 * 18432, 9216, 96, 0, 96, 96, 128, 0, 1.f, 0.f, 1);
  // A3 = M3 @ T3^H                                    -> slot 6
  cg(ws + OFF_M3, 98304, 1024, 0, ws + OFF_T3, 98304, 1024, 0, 1, 1024, 0,
     X + 6 * 18432, 9216, 96, 0, 96, 96, 1024, 1, 1.f, 0.f, 1);
  // R[n] = Qmat @ J[n]  (J[n] is contiguous 2048x128: bsj=1, bsr=128), batch 16
  cg(ws + OFF_Q, 196608, 2048, 0, J, 4194304, 1, 0, 128, 2048, 262144,
     ws + OFF_R, 196608, 128, 12288, 96, 128, 2048, 0, 1.f, 0.f, 16);
  // B3 = -TL @ R   (R[n,f,l]: bsj=128, bsq=12288, bsr=1, TD=128) -> slot 7
  cg(ws + OFF_TL, 196608, 2048, 0, ws + OFF_R, 196608, 128, 12288, 1, 128, 0,
     X + 7 * 18432, 9216, 96, 0, 96, 96, 2048, 0, -1.f, 0.f, 1);
  // C3 = C1                                           -> slot 8
  k_copy<<<(18432 + 255) / 256, 256, 0, stream>>>(X + 2 * 18432, X + 8 * 18432, 18432);

  // ---- stage 3: CNN head --------------------------------------------------
  k_conv<<<(12 * 48 * 48 + 255) / 256, 256, 0, stream>>>(X, c1w, c1b, ws + OFF_C1O, 18, 12, 96, 48);
  k_conv<<<(8 * 24 * 24 + 255) / 256, 256, 0, stream>>>(ws + OFF_C1O, c2w, c2b, ws + OFF_C2O, 12, 8, 48, 24);
  k_conv<<<(6 * 12 * 12 + 255) / 256, 256, 0, stream>>>(ws + OFF_C2O, c3w, c3b, ws + OFF_C3O, 8, 6, 24, 12);
  k_fc<<<(1000 + 255) / 256, 256, 0, stream>>>(ws + OFF_C3O, f1w, f1b, ws + OFF_F1, 864, 1000);
  k_fc<<<(700 + 255) / 256, 256, 0, stream>>>(ws + OFF_F1, f2w, f2b, ws + OFF_F2, 1000, 700);
  k_fc<<<1, 192, 0, stream>>>(ws + OFF_F2, f3w, f3b, (float*)d_out, 700, 192);
}